// SVD_Solver_23381801960249
// MI455X (gfx1250) — compile-verified
//
#include <hip/hip_runtime.h>

// Weighted Kabsch (per batch x joint rigid fit) for MI455X / gfx1250.
// B=4096, M=90 markers, J=31 joints. Memory-bound (~193 MB -> ~8.3us roofline).
// Per (b,j), all 23 reduction sums (SZX 3x3, SmZ, SmX, cnt, SwX, wsum, SwZ)
// are one 8x92 @ 92x4 GEMM executed as 23 V_WMMA_F32_16X16X4_F32 steps
// (K padded to 92 with zeroed LDS), two independent accumulator chains.

#define N_M 90
#define N_K 92          // K padded to multiple of 4
#define N_J 31

typedef __attribute__((ext_vector_type(2))) float v2f;
typedef __attribute__((ext_vector_type(8))) float v8f;

__device__ __forceinline__ void jacobi_rot(float A[3][3], float V[3][3],
                                           int p, int q) {
  float apq = A[p][q];
  if (fabsf(apq) < 1e-30f) return;
  float tau = (A[q][q] - A[p][p]) / (2.0f * apq);
  float t = ((tau >= 0.0f) ? 1.0f : -1.0f) /
            (fabsf(tau) + sqrtf(1.0f + tau * tau));
  float c = rsqrtf(1.0f + t * t);
  float s = t * c;
#pragma unroll
  for (int k = 0; k < 3; ++k) {
    float akp = A[k][p], akq = A[k][q];
    A[k][p] = c * akp - s * akq;
    A[k][q] = s * akp + c * akq;
  }
#pragma unroll
  for (int k = 0; k < 3; ++k) {
    float apk = A[p][k], aqk = A[q][k];
    A[p][k] = c * apk - s * aqk;
    A[q][k] = s * apk + c * aqk;
  }
#pragma unroll
  for (int k = 0; k < 3; ++k) {
    float vkp = V[k][p], vkq = V[k][q];
    V[k][p] = c * vkp - s * vkq;
    V[k][q] = s * vkp + c * vkq;
  }
}

__global__ __launch_bounds__(256) void kabsch_wmma_kernel(
    const float* __restrict__ Xg,  // [B, M, 3]
    const float* __restrict__ Wg,  // [B, M, J]
    const float* __restrict__ Zg,  // [B, M, J, 3]
    float* __restrict__ out) {     // [B, J, 3, 4]
  __shared__ float sZ[N_K * N_J * 3];  // 8556 (rows 90,91 zeroed)
  __shared__ float sM[N_K * N_J];      // 2852 : mask (w > 0.01)
  __shared__ float sWM[N_K * N_J];     // 2852 : w * mask
  __shared__ float sX[N_K * 3];        // 276  (rows 90,91 zeroed)
  __shared__ float sRes[N_J * 32];     // per-joint 8x4 WMMA result block

  const int b = blockIdx.x;
  const int tid = threadIdx.x;

  // ---- Stage this batch's slices into LDS. Z/X copied (b64 coalesced, all
  // slice byte offsets are multiples of 8); w is thresholded once into the
  // {m, w*m} pair; K dimension zero-padded to 92 so the WMMA loop is
  // completely guard-free. ----
  {
    const v2f* gz = (const v2f*)(Zg + (size_t)b * (N_M * N_J * 3));
    v2f* lz = (v2f*)sZ;
    for (int i = tid; i < (N_M * N_J * 3) / 2; i += 256) {
      __builtin_prefetch(&gz[i + 1024], 0, 0);  // speculative gfx1250 prefetch
      lz[i] = gz[i];
    }
    if (tid < (N_K - N_M) * N_J * 3) sZ[N_M * N_J * 3 + tid] = 0.0f;

    const float* gw = Wg + (size_t)b * (N_M * N_J);
    for (int i = tid; i < N_M * N_J; i += 256) {
      float wv = gw[i];
      float mv = (wv > 0.01f) ? 1.0f : 0.0f;
      sM[i] = mv;
      sWM[i] = wv * mv;
    }
    if (tid < (N_K - N_M) * N_J) {
      sM[N_M * N_J + tid] = 0.0f;
      sWM[N_M * N_J + tid] = 0.0f;
    }

    const v2f* gx = (const v2f*)(Xg + (size_t)b * (N_M * 3));
    v2f* lx = (v2f*)sX;
    for (int i = tid; i < (N_M * 3) / 2; i += 256) lx[i] = gx[i];
    if (tid < (N_K - N_M) * 3) sX[N_M * 3 + tid] = 0.0f;
  }
  __syncthreads();

  // ---- Per-lane constants for fragment construction (branchless). A rows:
  //  0..2 : m * Z[k,j,row]     3 : m     4 : w*m     5..7 : w*m * Z[k,j,row-5]
  // B cols: 0..2 : X[k,col]    3 : 1.0   else 0. ----
  const int lane = tid & 31;
  const int wv = tid >> 5;
  const int half = lane >> 4;
  const int lm = lane & 15;

  const bool rowAct = lm < 8;
  const bool rowOne = (lm == 3) || (lm == 4);            // multiply by 1, not Z
  const int zIdx = (lm < 3) ? lm : ((lm >= 5 && lm < 8) ? lm - 5 : 0);
  const float* coefBase = (lm >= 4) ? sWM : sM;          // rows 4..7 use w*m
  const bool colAct = lm < 3;
  const float colDflt = (lm == 3) ? 1.0f : 0.0f;
  const int xIdx = colAct ? lm : 0;

  // B fragments depend only on k: build once, keep in registers.
  // f32 A/B 16x4 layout: VGPR0 = {K0 | K2}, VGPR1 = {K1 | K3} per half-wave.
  v2f bfrag[23];
#pragma unroll
  for (int s = 0; s < 23; ++s) {
    const int kb = 4 * s + 2 * half;
    float x0 = sX[kb * 3 + xIdx];
    float x1 = sX[(kb + 1) * 3 + xIdx];
    v2f t;
    t.x = colAct ? x0 : colDflt;
    t.y = colAct ? x1 : colDflt;
    bfrag[s] = t;
  }

  for (int j = wv; j < N_J; j += 8) {  // wave-uniform loop: EXEC stays full
    v8f acc0 = {}, acc1 = {};
#pragma unroll
    for (int s = 0; s < 23; ++s) {
      const int ka = 4 * s + 2 * half;
      const int i0 = ka * N_J + j;
      const int i1 = i0 + N_J;
      float c0 = coefBase[i0];
      float c1 = coefBase[i1];
      float z0 = sZ[i0 * 3 + zIdx];
      float z1 = sZ[i1 * 3 + zIdx];
      float e0 = c0 * (rowOne ? 1.0f : z0);
      float e1 = c1 * (rowOne ? 1.0f : z1);
      v2f av;
      av.x = rowAct ? e0 : 0.0f;
      av.y = rowAct ? e1 : 0.0f;
      if (s & 1)
        acc1 = __builtin_amdgcn_wmma_f32_16x16x4_f32(
            false, av, false, bfrag[s], (short)0, acc1, false, false);
      else
        acc0 = __builtin_amdgcn_wmma_f32_16x16x4_f32(
            false, av, false, bfrag[s], (short)0, acc0, false, false);
    }
    // D layout: VGPR r holds row r (lanes 0-15), lane index = column.
    if (lane < 4) {
#pragma unroll
      for (int r = 0; r < 8; ++r)
        sRes[j * 32 + r * 4 + lane] = acc0[r] + acc1[r];
    }
  }
  __syncthreads();

  // ---- Per-joint 3x3 SVD + rigid transform (wave 0, one lane per joint) ----
  if (tid < N_J) {
    const int j = tid;
    float r32[32];
#pragma unroll
    for (int i = 0; i < 32; ++i) r32[i] = sRes[j * 32 + i];

    const float cnt = r32[15];   // row3,col3 : sum(m)
    const float wsum = r32[19];  // row4,col3 : sum(w*m)
    const float inv = 1.0f / fmaxf(wsum, 1e-20f);
    float Zb[3], Xb[3];
#pragma unroll
    for (int a = 0; a < 3; ++a) {
      Zb[a] = r32[(5 + a) * 4 + 3] * inv;  // SwZ / wsum
      Xb[a] = r32[16 + a] * inv;           // SwX / wsum
    }
    // S = SZX - Zbar*SmX^T - SmZ*Xbar^T + cnt*Zbar*Xbar^T
    float S[3][3];
#pragma unroll
    for (int a = 0; a < 3; ++a)
#pragma unroll
      for (int c = 0; c < 3; ++c)
        S[a][c] = r32[a * 4 + c] - Zb[a] * r32[12 + c] -
                  Xb[c] * r32[a * 4 + 3] + cnt * Zb[a] * Xb[c];

    // Jacobi eigen-solve of A = S^T S  ->  V, sigma^2
    float A[3][3], V[3][3];
#pragma unroll
    for (int i = 0; i < 3; ++i)
#pragma unroll
      for (int c = 0; c < 3; ++c) {
        A[i][c] = S[0][i] * S[0][c] + S[1][i] * S[1][c] + S[2][i] * S[2][c];
        V[i][c] = (i == c) ? 1.0f : 0.0f;
      }
#pragma unroll
    for (int sweep = 0; sweep < 4; ++sweep) {
      jacobi_rot(A, V, 0, 1);
      jacobi_rot(A, V, 0, 2);
      jacobi_rot(A, V, 1, 2);
    }
    float eig[3] = {A[0][0], A[1][1], A[2][2]};
#define SORT_PAIR(i0, i1)                                              \
  if (eig[i0] < eig[i1]) {                                             \
    float te = eig[i0]; eig[i0] = eig[i1]; eig[i1] = te;               \
    for (int k = 0; k < 3; ++k) {                                      \
      float tv = V[k][i0]; V[k][i0] = V[k][i1]; V[k][i1] = tv;         \
    }                                                                  \
  }
    SORT_PAIR(0, 1)
    SORT_PAIR(1, 2)
    SORT_PAIR(0, 1)
#undef SORT_PAIR

    // U columns: u_i = normalize(S v_i)  (descending sigma order)
    float U[3][3];
#pragma unroll
    for (int i = 0; i < 3; ++i) {
      float u0 = S[0][0] * V[0][i] + S[0][1] * V[1][i] + S[0][2] * V[2][i];
      float u1 = S[1][0] * V[0][i] + S[1][1] * V[1][i] + S[1][2] * V[2][i];
      float u2 = S[2][0] * V[0][i] + S[2][1] * V[1][i] + S[2][2] * V[2][i];
      float nin = 1.0f / fmaxf(sqrtf(u0 * u0 + u1 * u1 + u2 * u2), 1e-20f);
      U[0][i] = u0 * nin;
      U[1][i] = u1 * nin;
      U[2][i] = u2 * nin;
    }
    // det(U Vh) == sign(det S) since all sigma >= 0.
    float detS = S[0][0] * (S[1][1] * S[2][2] - S[1][2] * S[2][1]) -
                 S[0][1] * (S[1][0] * S[2][2] - S[1][2] * S[2][0]) +
                 S[0][2] * (S[1][0] * S[2][1] - S[1][1] * S[2][0]);
    float dflip = (detS < 0.0f) ? -1.0f : 1.0f;

    // R = V diag(1,1,det) U^T ; t = Xbar - R Zbar ; emit [R | t] (3x4)
    float* o = out + ((size_t)b * N_J + j) * 12;
#pragma unroll
    for (int a = 0; a < 3; ++a) {
      float r0 = V[a][0] * U[0][0] + V[a][1] * U[0][1] + dflip * V[a][2] * U[0][2];
      float r1 = V[a][0] * U[1][0] + V[a][1] * U[1][1] + dflip * V[a][2] * U[1][2];
      float r2 = V[a][0] * U[2][0] + V[a][1] * U[2][1] + dflip * V[a][2] * U[2][2];
      float ta = Xb[a] - (r0 * Zb[0] + r1 * Zb[1] + r2 * Zb[2]);
      o[a * 4 + 0] = r0;
      o[a * 4 + 1] = r1;
      o[a * 4 + 2] = r2;
      o[a * 4 + 3] = ta;
    }
  }
}

extern "C" void kernel_launch(void* const* d_in, const int* in_sizes, int n_in,
                              void* d_out, int out_size, void* d_ws,
                              size_t ws_size, hipStream_t stream) {
  (void)n_in;
  (void)out_size;
  (void)d_ws;
  (void)ws_size;
  const float* X = (const float*)d_in[0];  // [B,M,3]
  const float* W = (const float*)d_in[1];  // [B,M,J]
  const float* Z = (const float*)d_in[2];  // [B,M,J,3]
  float* out = (float*)d_out;              // [B,J,3,4]
  const int B = in_sizes[0] / (N_M * 3);   // 4096
  hipLaunchKernelGGL(kabsch_wmma_kernel, dim3(B), dim3(256), 0, stream, X, W,
                     Z, out);
}